// PointPillarsScatter_46050639347746
// MI455X (gfx1250) — compile-verified
//
#include <hip/hip_runtime.h>
#include <stdint.h>

// PointPillarsScatter for MI455X (gfx1250).
// Zero-FLOP op => pure data movement. Strategy: invert scatter into a gather so
// the 215 MB output is written exactly once, coalesced, and stage per-point
// 512B feature vectors HBM -> LDS with gfx1250 async-to-LDS DMA instructions
// (ASYNCcnt-tracked), then do an LDS-tiled transpose to the [B,C,Y,X] layout.

static constexpr int kNY = 162;
static constexpr int kNX = 162;
static constexpr int kC  = 128;
static constexpr int kTileX = 32;
static constexpr int kTilesPerRow = (kNX + kTileX - 1) / kTileX;  // 6
static constexpr int kSlotStride  = kC + 4;  // 132 floats = 528B: 16B aligned, 2-way-conflict transpose reads

typedef float v4f __attribute__((ext_vector_type(4)));  // native vector for nontemporal builtins

// ---------------- small utility kernels ----------------

__global__ void zero_i4_kernel(int4* __restrict__ p, long long n4) {
  long long i = (long long)blockIdx.x * blockDim.x + threadIdx.x;
  long long stride = (long long)gridDim.x * blockDim.x;
  int4 z = {0, 0, 0, 0};
  for (; i < n4; i += stride) p[i] = z;
}

__global__ void zero_f4_kernel(v4f* __restrict__ p, long long n4) {
  long long i = (long long)blockIdx.x * blockDim.x + threadIdx.x;
  long long stride = (long long)gridDim.x * blockDim.x;
  v4f z = {0.f, 0.f, 0.f, 0.f};
  for (; i < n4; i += stride) __builtin_nontemporal_store(z, &p[i]);
}

// idxmap[b, y, x] = n + 1 for each point n (cells are unique per batch).
__global__ void build_index_kernel(const int4* __restrict__ coors,
                                   int* __restrict__ idxmap, int n) {
  int i = blockIdx.x * blockDim.x + threadIdx.x;
  if (i >= n) return;
  int4 cc = coors[i];  // [batch, z, y, x]
  int flat = (cc.x * kNY + cc.z) * kNX + cc.w;
  idxmap[flat] = i + 1;
}

// ---------------- main gather/transpose kernel ----------------
// One 256-thread block (8 wave32) per (b, y, 32-wide x tile).
//  phase 1: wave 0 loads 32 cell indices (coalesced).
//  phase 2: each wave stages 4 cells: non-empty -> async HBM->LDS DMA of the
//           512B feature vector (global_load_async_to_lds_b128, 16B/lane);
//           empty -> ds_store_b128 zeros. (Disjoint LDS ranges, so the two
//           counter domains never race on the same bytes.)
//  phase 3: s_wait_asynccnt 0 + barrier, then transposed coalesced
//           nontemporal writes: out[b, c, y, x0+lane] = tile[lane][c].
__global__ void __launch_bounds__(256)
pillars_gather_transpose_kernel(const float* __restrict__ vf,
                                const int* __restrict__ idxmap,
                                float* __restrict__ out) {
  __shared__ __align__(16) float tile[kTileX * kSlotStride];  // 16896 B
  __shared__ int s_idx[kTileX];

  const int tid  = threadIdx.x;
  const int lane = tid & 31;
  const int wid  = tid >> 5;  // 0..7

  int blk = blockIdx.x;
  const int t  = blk % kTilesPerRow;  blk /= kTilesPerRow;
  const int y  = blk % kNY;
  const int b  = blk / kNY;
  const int x0 = t * kTileX;

  if (wid == 0) {
    int xx = x0 + lane;
    s_idx[lane] = (xx < kNX) ? idxmap[(b * kNY + y) * kNX + xx] : 0;
  }
  __syncthreads();

  const uint32_t lds_base = (uint32_t)(uintptr_t)(&tile[0]);  // low 32b of generic addr == LDS offset
  for (int j = wid * 4; j < wid * 4 + 4; ++j) {
    const int idxv = s_idx[j];  // wave-uniform
    if (idxv > 0) {
      const float* g = vf + (size_t)(idxv - 1) * kC + lane * 4;  // 16B per lane, 512B per point
      uint32_t l = lds_base + (uint32_t)(j * kSlotStride * 4) + (uint32_t)(lane * 16);
      asm volatile("global_load_async_to_lds_b128 %0, %1, off"
                   :: "v"(l), "v"(g)
                   : "memory");
    } else {
      float4 z = {0.f, 0.f, 0.f, 0.f};
      *reinterpret_cast<float4*>(&tile[j * kSlotStride + lane * 4]) = z;
    }
  }
  asm volatile("s_wait_asynccnt 0" ::: "memory");  // own DMA done (LDS visible WGP-wide)
  __syncthreads();                                 // everyone's DMA/zeros done

  const int xx = x0 + lane;
  if (xx < kNX) {
    size_t orow = (((size_t)b * kC + wid * 16) * kNY + y) * (size_t)kNX + xx;
#pragma unroll
    for (int k = 0; k < 16; ++k) {  // wave covers channels [wid*16, wid*16+16)
      float v = tile[lane * kSlotStride + wid * 16 + k];
      __builtin_nontemporal_store(v, out + orow);  // 128B coalesced per wave
      orow += (size_t)kNY * kNX;
    }
  }
}

// ---------------- fallback: direct scatter (if d_ws too small) ----------------
__global__ void direct_scatter_kernel(const float* __restrict__ vf,
                                      const int4* __restrict__ coors,
                                      float* __restrict__ out, int n) {
  int gid = blockIdx.x * blockDim.x + threadIdx.x;
  int np = gid >> 5;
  int q  = gid & 31;
  if (np >= n) return;
  int4 cc = coors[np];
  const size_t plane = (size_t)kNY * kNX;
  size_t base = (size_t)cc.x * kC * plane + (size_t)cc.z * kNX + (size_t)cc.w;
  float4 f = reinterpret_cast<const float4*>(vf)[(size_t)np * 32 + q];
  float vals[4] = {f.x, f.y, f.z, f.w};
#pragma unroll
  for (int k = 0; k < 4; ++k)
    out[base + (size_t)(4 * q + k) * plane] = vals[k];
}

// ---------------- launcher ----------------
extern "C" void kernel_launch(void* const* d_in, const int* in_sizes, int n_in,
                              void* d_out, int out_size, void* d_ws, size_t ws_size,
                              hipStream_t stream) {
  const float* vf   = (const float*)d_in[0];
  const int*   coor = (const int*)d_in[1];
  float* out = (float*)d_out;

  const int N = in_sizes[0] / kC;                    // 208000
  const int B = out_size / (kC * kNY * kNX);         // 16
  const size_t mapElems = (size_t)B * kNY * kNX;     // 419904 (mult of 4)

  const int threads = 256;

  if (d_ws && ws_size >= mapElems * sizeof(int)) {
    int* idxmap = (int*)d_ws;

    long long n4 = (long long)(mapElems / 4);
    int gz = (int)((n4 + threads - 1) / threads);
    zero_i4_kernel<<<gz, threads, 0, stream>>>((int4*)idxmap, n4);

    int gi = (N + threads - 1) / threads;
    build_index_kernel<<<gi, threads, 0, stream>>>((const int4*)coor, idxmap, N);

    int gb = B * kNY * kTilesPerRow;  // 15552 blocks
    pillars_gather_transpose_kernel<<<gb, threads, 0, stream>>>(vf, idxmap, out);
  } else {
    // Fallback: zero canvas then strided scatter.
    long long n4 = (long long)out_size / 4;
    int gz = (int)((n4 + threads - 1) / threads);
    zero_f4_kernel<<<gz, threads, 0, stream>>>((v4f*)out, n4);

    long long tot = (long long)N * 32;
    int gs = (int)((tot + threads - 1) / threads);
    direct_scatter_kernel<<<gs, threads, 0, stream>>>(vf, (const int4*)coor, out, N);
  }
}